// TrafficCCNN_30648886624259
// MI455X (gfx1250) — compile-verified
//
#include <hip/hip_runtime.h>

typedef __attribute__((ext_vector_type(2))) float v2f;
typedef __attribute__((ext_vector_type(8))) float v8f;

static constexpr int kB = 16, kW = 12, kN0 = 8192, kN1 = 32768, kN2 = 16384;
static constexpr int kH = 64, kPred = 12;
static constexpr int E0 = 65536, E1 = 65536, E2 = 49152;

// ---------------------------------------------------------------------------
// Small projection: C[N x 64] = F[N x K] @ Wsub[K x 64] + bias[64]
__global__ __launch_bounds__(256) void proj_small(
    const float* __restrict__ F, int K, const float* __restrict__ Wsub,
    const float* __restrict__ bias, float* __restrict__ C) {
  int t = blockIdx.x * 256 + threadIdx.x;
  int row = t >> 6, col = t & 63;
  float acc = bias[col];
  const float* fr = F + (size_t)row * K;
  for (int k = 0; k < K; ++k) acc += fr[k] * Wsub[k * 64 + col];
  C[(size_t)row * 64 + col] = acc;
}

// ---------------------------------------------------------------------------
// Temporal encoder: conv1d(1->64,k3,p1)+relu -> conv1d(64->64,k3,p1)+relu -> mean(t)
// One 64-thread block per (b, node); thread = output channel.
__global__ __launch_bounds__(64) void tempenc(
    const float* __restrict__ x, const float* __restrict__ w1,
    const float* __restrict__ b1, const float* __restrict__ w2,
    const float* __restrict__ b2, float* __restrict__ h) {
  __shared__ float h1[64 * 12];
  int bn = blockIdx.x;                 // b*N0 + n
  int b = bn >> 13, n = bn & (kN0 - 1);
  int ch = threadIdx.x;
  float xv[14];
  xv[0] = 0.f; xv[13] = 0.f;
  const float* xp = x + (size_t)b * kW * kN0 + n;
  #pragma unroll
  for (int t = 0; t < 12; ++t) xv[t + 1] = xp[(size_t)t * kN0];
  float wa = w1[ch * 3 + 0], wb = w1[ch * 3 + 1], wc = w1[ch * 3 + 2], bb = b1[ch];
  #pragma unroll
  for (int t = 0; t < 12; ++t) {
    float v = wa * xv[t] + wb * xv[t + 1] + wc * xv[t + 2] + bb;
    h1[ch * 12 + t] = v > 0.f ? v : 0.f;
  }
  __syncthreads();
  float acc[12];
  float cb = b2[ch];
  #pragma unroll
  for (int t = 0; t < 12; ++t) acc[t] = cb;
  for (int ic = 0; ic < 64; ++ic) {
    float hv[14];
    hv[0] = 0.f; hv[13] = 0.f;
    #pragma unroll
    for (int t = 0; t < 12; ++t) hv[t + 1] = h1[ic * 12 + t];
    const float* wp = w2 + (size_t)(ch * 64 + ic) * 3;
    float u0 = wp[0], u1 = wp[1], u2 = wp[2];
    #pragma unroll
    for (int t = 0; t < 12; ++t) acc[t] += u0 * hv[t] + u1 * hv[t + 1] + u2 * hv[t + 2];
  }
  float s = 0.f;
  #pragma unroll
  for (int t = 0; t < 12; ++t) s += acc[t] > 0.f ? acc[t] : 0.f;
  h[(size_t)bn * 64 + ch] = s * (1.0f / 12.0f);
}

// ---------------------------------------------------------------------------
// WMMA fp32 GEMM: C[N x 64] = Aeff[N x 64] @ W[64 x 64] (+bias) (+Dadd[row&dmask]) (relu?)
// Aeff = (COMPOSE ? relu(A1[r]) + relu(A2[r & amask]) : A1[r])
// One wave -> 16x64 C tile. W staged in LDS in fragment-paired order so every
// B fragment is one aligned ds_load_b64; A preloaded to registers; inner loop is
// 64 straight-line V_WMMA_F32_16X16X4_F32.
// NOTE: C may alias A1 (each wave reads only its own 16 A rows before writing them).
template <bool COMPOSE, bool HAS_BIAS, bool HAS_DADD, bool RELU>
__global__ __launch_bounds__(256) void gemm64(
    const float* A1, const float* A2, unsigned amask,
    const float* __restrict__ W, const float* __restrict__ bias,
    const float* __restrict__ Dadd, unsigned dmask, float* C) {
  __shared__ float wlds[64 * 64];
  // wlds[(kstep*128 + half*64 + col)*2 + j] = W[(kstep*4 + half*2 + j)*64 + col]
  for (int i = threadIdx.x; i < 4096; i += 256) {
    int k = i >> 6, col = i & 63;
    wlds[(((k >> 2) * 128) + (((k >> 1) & 1) * 64) + col) * 2 + (k & 1)] = W[i];
  }
  __syncthreads();
  int wid = threadIdx.x >> 5, lane = threadIdx.x & 31;
  int half = lane >> 4, l = lane & 15;
  int row0 = (blockIdx.x * 8 + wid) * 16;

  // Preload this lane's 32 A elements (pairs at K = 4s + 2*half + {0,1}).
  const float* a1p = A1 + (size_t)(row0 + l) * 64 + 2 * half;
  float areg[32];
  #pragma unroll
  for (int s = 0; s < 16; ++s) {
    areg[2 * s]     = a1p[4 * s];
    areg[2 * s + 1] = a1p[4 * s + 1];
  }
  if (COMPOSE) {
    const float* a2p = A2 + (size_t)((unsigned)(row0 + l) & amask) * 64 + 2 * half;
    #pragma unroll
    for (int s = 0; s < 16; ++s) {
      areg[2 * s]     = fmaxf(areg[2 * s], 0.f)     + fmaxf(a2p[4 * s], 0.f);
      areg[2 * s + 1] = fmaxf(areg[2 * s + 1], 0.f) + fmaxf(a2p[4 * s + 1], 0.f);
    }
  }

  v8f acc[4] = {};
  const float* wl = wlds + ((half * 64 + l) * 2);  // + s*256 + n*32
  #pragma unroll
  for (int s = 0; s < 16; ++s) {
    v2f af = {areg[2 * s], areg[2 * s + 1]};
    #pragma unroll
    for (int n = 0; n < 4; ++n) {
      v2f bf = *(const v2f*)(wl + s * 256 + n * 32);
      acc[n] = __builtin_amdgcn_wmma_f32_16x16x4_f32(false, af, false, bf,
                                                     (short)0, acc[n], false, false);
    }
  }

  #pragma unroll
  for (int n = 0; n < 4; ++n) {
    #pragma unroll
    for (int r = 0; r < 8; ++r) {
      int row = row0 + r + 8 * half;
      int col = n * 16 + l;
      float v = acc[n][r];
      if (HAS_BIAS) v += bias[col];
      if (HAS_DADD) v += Dadd[(size_t)((unsigned)row & dmask) * 64 + col];
      if (RELU) v = fmaxf(v, 0.f);
      C[(size_t)row * 64 + col] = v;
    }
  }
}

// ---------------------------------------------------------------------------
// Row dots: s1[r] = Ceff[r]·v1 (and s2[r] = Ceff[r]·v2 if v2), wave per row.
__global__ __launch_bounds__(256) void rowdot2(
    const float* __restrict__ C1, const float* C2, unsigned amask,
    const float* __restrict__ v1, const float* v2,
    float* __restrict__ s1, float* s2) {
  int row = blockIdx.x * 8 + (threadIdx.x >> 5);
  int lane = threadIdx.x & 31;
  const float* r1 = C1 + (size_t)row * 64;
  float a = r1[lane], b = r1[lane + 32];
  if (C2) {
    const float* r2 = C2 + (size_t)((unsigned)row & amask) * 64;
    a = fmaxf(a, 0.f) + fmaxf(r2[lane], 0.f);
    b = fmaxf(b, 0.f) + fmaxf(r2[lane + 32], 0.f);
  }
  float t1 = a * v1[lane] + b * v1[lane + 32];
  for (int m = 16; m; m >>= 1) t1 += __shfl_xor(t1, m, 32);
  if (lane == 0) s1[row] = t1;
  if (v2) {
    float t2 = a * v2[lane] + b * v2[lane + 32];
    for (int m = 16; m; m >>= 1) t2 += __shfl_xor(t2, m, 32);
    if (lane == 0) s2[row] = t2;
  }
}

// 64x64 matvec (one block): out[c] = sum_k W[k][c] * v[k]
__global__ __launch_bounds__(64) void matvec64(
    const float* __restrict__ W, const float* __restrict__ v, float* __restrict__ out) {
  int c = threadIdx.x;
  float acc = 0.f;
  for (int k = 0; k < 64; ++k) acc += W[k * 64 + c] * v[k];
  out[c] = acc;
}

// ---------------------------------------------------------------------------
// Edge pass 1: e_j = leaky_relu(s1[b*str1+i1[i]] + s2[b*str2+i2[i]]);
// store e; rs[b*strr+ir[i]] += e.   (thread per edge; j = b*E + i)
__global__ __launch_bounds__(256) void epass1(
    const float* __restrict__ s1, const int* __restrict__ i1, int str1,
    const float* __restrict__ s2, const int* __restrict__ i2, int str2,
    float* __restrict__ rs, const int* __restrict__ ir, int strr,
    float* __restrict__ ebuf, int E) {
  int j = blockIdx.x * 256 + threadIdx.x;
  int b = j / E;
  int i = j - b * E;
  float v = s1[(size_t)b * str1 + i1[i]] + s2[(size_t)b * str2 + i2[i]];
  float e = v > 0.f ? v : 0.2f * v;
  ebuf[j] = e;
  atomicAdd(rs + (size_t)b * strr + ir[i], e);
}

// Edge pass 2: acc[b*strr+ir[i]][:] += (e / rs_guarded) * msg[b*strm+im[i]][:]
// wave per edge, lane handles 2 columns.
__global__ __launch_bounds__(256) void epass2(
    const float* __restrict__ ebuf, const float* __restrict__ rs,
    const int* __restrict__ ir, int strr,
    const float* __restrict__ msg, const int* __restrict__ im, int strm,
    float* __restrict__ acc, int E) {
  int wg = blockIdx.x * 8 + (threadIdx.x >> 5);
  int lane = threadIdx.x & 31;
  int b = wg / E;
  int i = wg - b * E;
  float e = ebuf[wg];
  int r = ir[i], m = im[i];
  float rv = rs[(size_t)b * strr + r];
  float att = e / (rv == 0.0f ? 1.0f : rv);
  const float* mrow = msg + ((size_t)b * strm + m) * 64;
  float* arow = acc + ((size_t)b * strr + r) * 64;
  atomicAdd(arow + lane, att * mrow[lane]);
  atomicAdd(arow + lane + 32, att * mrow[lane + 32]);
}

// ---------------------------------------------------------------------------
// Final decoder stage: out[b][p][n] = z[b,n,:]·dec2_w[:,p] + dec2_b[p]
__global__ __launch_bounds__(256) void decoder2(
    const float* __restrict__ z, const float* __restrict__ w,
    const float* __restrict__ bias, float* __restrict__ out) {
  int t = blockIdx.x * 256 + threadIdx.x;   // b*N0 + n
  int b = t >> 13, n = t & (kN0 - 1);
  const float* zr = z + (size_t)t * 64;
  float o[kPred];
  #pragma unroll
  for (int p = 0; p < kPred; ++p) o[p] = bias[p];
  for (int c = 0; c < 64; ++c) {
    float zv = zr[c];
    #pragma unroll
    for (int p = 0; p < kPred; ++p) o[p] += zv * w[c * kPred + p];
  }
  float* op = out + (size_t)b * kPred * kN0 + n;
  #pragma unroll
  for (int p = 0; p < kPred; ++p) op[(size_t)p * kN0] = o[p];
}

// ---------------------------------------------------------------------------
extern "C" void kernel_launch(void* const* d_in, const int* in_sizes, int n_in,
                              void* d_out, int out_size, void* d_ws, size_t ws_size,
                              hipStream_t stream) {
  (void)in_sizes; (void)n_in; (void)out_size; (void)ws_size;
  const float* x       = (const float*)d_in[0];
  const float* nodef   = (const float*)d_in[1];
  const float* edgef   = (const float*)d_in[2];
  const float* facef   = (const float*)d_in[3];
  const float* c1w     = (const float*)d_in[4];
  const float* c1b     = (const float*)d_in[5];
  const float* c2w     = (const float*)d_in[6];
  const float* c2b     = (const float*)d_in[7];
  const float* npw     = (const float*)d_in[8];
  const float* npb     = (const float*)d_in[9];
  const float* epw     = (const float*)d_in[10];
  const float* epb     = (const float*)d_in[11];
  const float* fpw     = (const float*)d_in[12];
  const float* fpb     = (const float*)d_in[13];
  const float* hbs_w   = (const float*)d_in[14];
  const float* hbs_a   = (const float*)d_in[15];
  const float* hws     = (const float*)d_in[16];
  const float* hwt     = (const float*)d_in[17];
  const float* hba     = (const float*)d_in[18];
  const float* dec1w   = (const float*)d_in[19];
  const float* dec1b   = (const float*)d_in[20];
  const float* dec2w   = (const float*)d_in[21];
  const float* dec2b   = (const float*)d_in[22];
  const int* a0r = (const int*)d_in[23];
  const int* a0c = (const int*)d_in[24];
  const int* b1r = (const int*)d_in[29];
  const int* b1c = (const int*)d_in[30];
  const int* b2r = (const int*)d_in[31];
  const int* b2c = (const int*)d_in[32];
  float* out = (float*)d_out;

  // ---- workspace arena (floats) ----
  float* ws = (float*)d_ws;
  size_t o = 0;
  auto alloc = [&](size_t n) { float* p = ws + o; o += n; return p; };
  const size_t NB1 = (size_t)kB * kN1 * 64;   // 33.5M fl
  const size_t NM  = (size_t)kB * kN0 * 64;   // 8.39M fl
  float* BIG  = alloc(NB1);   // acc01 -> sm2 (in place)
  float* M1   = alloc(NM);    // h -> m1 -> m2 -> z
  float* M2   = alloc(NM);    // x0 -> accy00
  float* M3   = alloc(NM);    // tm0 -> accy10
  float* M4   = alloc(NM);    // acc00
  float* M5   = alloc(NM);    // acc10
  float* sm0    = alloc((size_t)kN1 * 64);
  float* acc21  = alloc((size_t)kN1 * 64);
  float* edgein = alloc((size_t)kN1 * 64);
  float* facein = alloc((size_t)kN2 * 64);
  float* sm1    = alloc((size_t)kN2 * 64);
  float* tm1    = alloc((size_t)kN1 * 64);
  float* nfproj = alloc((size_t)kN0 * 64);
  float* psm0 = alloc(kN1); float* qsm0 = alloc(kN1);
  float* psm1 = alloc(kN2); float* ptm1 = alloc(kN1);
  float* pm1r = alloc((size_t)kB * kN0); float* pm1c = alloc((size_t)kB * kN0);
  float* ptm0 = alloc((size_t)kB * kN0); float* qtm0 = alloc((size_t)kB * kN0);
  float* pm2r = alloc((size_t)kB * kN0); float* pm2c = alloc((size_t)kB * kN0);
  float* ptm2 = alloc((size_t)kB * kN0);
  float* psm2 = alloc((size_t)kB * kN1);
  float* rs00 = alloc((size_t)kB * kN0); float* rs10 = alloc((size_t)kB * kN0);
  float* rsy00 = alloc((size_t)kB * kN0); float* rsy10 = alloc((size_t)kB * kN0);
  float* rs01 = alloc((size_t)kB * kN1); float* rs21 = alloc(kN1);
  float* wvec = alloc(64);
  float* ebuf = alloc((size_t)kB * E0);

  const unsigned FULL = 0xFFFFFFFFu;
  const int NBN0 = kB * kN0;           // 131072
  const int NBN1 = kB * kN1;           // 524288

  auto gemm_plain   = gemm64<false, false, false, false>;
  auto gemm_dadd    = gemm64<false, false, true,  false>;
  auto gemm_comp    = gemm64<true,  false, false, false>;
  auto gemm_comp_br = gemm64<true,  true,  false, true>;

  // ================= Precompute (batch-independent) =================
  proj_small<<<kN0 * 64 / 256, 256, 0, stream>>>(nodef, 7, npw + 64 * 64, npb, nfproj);
  proj_small<<<kN1 * 64 / 256, 256, 0, stream>>>(edgef, 4, epw + 64 * 64, epb, edgein);
  proj_small<<<kN2 * 64 / 256, 256, 0, stream>>>(facef, 3, fpw + 64 * 64, fpb, facein);

  gemm_plain<<<kN1 / 128, 256, 0, stream>>>(edgein, nullptr, 0, hws + 0, nullptr, nullptr, 0, sm0);
  rowdot2<<<kN1 / 8, 256, 0, stream>>>(sm0, nullptr, 0, hba + 0, hba + 64, psm0, qsm0);
  gemm_plain<<<kN2 / 128, 256, 0, stream>>>(facein, nullptr, 0, hws + 4096, nullptr, nullptr, 0, sm1);
  rowdot2<<<kN2 / 8, 256, 0, stream>>>(sm1, nullptr, 0, hba + 128, nullptr, psm1, nullptr);
  gemm_plain<<<kN1 / 128, 256, 0, stream>>>(edgein, nullptr, 0, hwt + 4096, nullptr, nullptr, 0, tm1);
  rowdot2<<<kN1 / 8, 256, 0, stream>>>(tm1, nullptr, 0, hba + 192, nullptr, ptm1, nullptr);
  matvec64<<<1, 64, 0, stream>>>(hwt + 8192, hba + 320, wvec);

  // HBNS1 e-side -> acc21 (x21 pre-relu), batch independent
  hipMemsetAsync(acc21, 0, (size_t)kN1 * 64 * 4, stream);
  hipMemsetAsync(rs21, 0, (size_t)kN1 * 4, stream);
  epass1<<<E2 / 256, 256, 0, stream>>>(psm1, b2c, 0, ptm1, b2r, 0, rs21, b2r, 0, ebuf, E2);
  epass2<<<E2 / 8, 256, 0, stream>>>(ebuf, rs21, b2r, 0, sm1, b2c, 0, acc21, E2);

  // ================= Batched stages =================
  tempenc<<<kB * kN0, 64, 0, stream>>>(x, c1w, c1b, c2w, c2b, M1);                        // h
  gemm_dadd<<<NBN0 / 128, 256, 0, stream>>>(M1, nullptr, 0, npw, nullptr, nfproj, kN0 - 1, M2);  // x0
  gemm_plain<<<NBN0 / 128, 256, 0, stream>>>(M2, nullptr, 0, hbs_w, nullptr, nullptr, 0, M1);    // m1
  rowdot2<<<NBN0 / 8, 256, 0, stream>>>(M1, nullptr, 0, hbs_a + 0, hbs_a + 64, pm1r, pm1c);
  gemm_plain<<<NBN0 / 128, 256, 0, stream>>>(M2, nullptr, 0, hwt, nullptr, nullptr, 0, M3);      // tm0
  rowdot2<<<NBN0 / 8, 256, 0, stream>>>(M3, nullptr, 0, hba + 64, hba + 0, ptm0, qtm0);

  // HBS level 1 -> acc00 (M4)
  hipMemsetAsync(M4, 0, NM * 4, stream);
  hipMemsetAsync(rs00, 0, (size_t)NBN0 * 4, stream);
  epass1<<<kB * E0 / 256, 256, 0, stream>>>(pm1r, a0r, kN0, pm1c, a0c, kN0, rs00, a0r, kN0, ebuf, E0);
  epass2<<<kB * E0 / 8, 256, 0, stream>>>(ebuf, rs00, a0r, kN0, M1, a0c, kN0, M4, E0);

  // HBNS0 e-side -> acc10 (M5)
  hipMemsetAsync(M5, 0, NM * 4, stream);
  hipMemsetAsync(rs10, 0, (size_t)NBN0 * 4, stream);
  epass1<<<kB * E1 / 256, 256, 0, stream>>>(psm0, b1c, 0, ptm0, b1r, kN0, rs10, b1r, kN0, ebuf, E1);
  epass2<<<kB * E1 / 8, 256, 0, stream>>>(ebuf, rs10, b1r, kN0, sm0, b1c, 0, M5, E1);

  // HBNS0 f-side -> acc01 (BIG)
  hipMemsetAsync(BIG, 0, NB1 * 4, stream);
  hipMemsetAsync(rs01, 0, (size_t)NBN1 * 4, stream);
  epass1<<<kB * E1 / 256, 256, 0, stream>>>(qtm0, b1r, kN0, qsm0, b1c, 0, rs01, b1c, kN1, ebuf, E1);
  epass2<<<kB * E1 / 8, 256, 0, stream>>>(ebuf, rs01, b1c, kN1, M3, b1r, kN0, BIG, E1);

  // ptm2 = (relu(acc00)+relu(acc10)) · wvec    (x0_l1 scalar, composed)
  rowdot2<<<NBN0 / 8, 256, 0, stream>>>(M4, M5, FULL, wvec, nullptr, ptm2, nullptr);

  // sm2 = x1_l1 @ hbns_ws[2], x1_l1 = relu(acc01)+relu(acc21[mod N1]) ; in-place over BIG
  gemm_comp<<<NBN1 / 128, 256, 0, stream>>>(BIG, acc21, (unsigned)(kN1 - 1), hws + 8192,
                                            nullptr, nullptr, 0, BIG);
  rowdot2<<<NBN1 / 8, 256, 0, stream>>>(BIG, nullptr, 0, hba + 256, nullptr, psm2, nullptr);

  // m2 = x0_l1 @ hbs_w[1], x0_l1 = relu(acc00)+relu(acc10)
  gemm_comp<<<NBN0 / 128, 256, 0, stream>>>(M4, M5, FULL, hbs_w + 4096, nullptr, nullptr, 0, M1);
  rowdot2<<<NBN0 / 8, 256, 0, stream>>>(M1, nullptr, 0, hbs_a + 128, hbs_a + 192, pm2r, pm2c);

  // HBS level 2 -> accy00 (M2)
  hipMemsetAsync(M2, 0, NM * 4, stream);
  hipMemsetAsync(rsy00, 0, (size_t)NBN0 * 4, stream);
  epass1<<<kB * E0 / 256, 256, 0, stream>>>(pm2r, a0r, kN0, pm2c, a0c, kN0, rsy00, a0r, kN0, ebuf, E0);
  epass2<<<kB * E0 / 8, 256, 0, stream>>>(ebuf, rsy00, a0r, kN0, M1, a0c, kN0, M2, E0);

  // HBNS2 e-side -> accy10 (M3)
  hipMemsetAsync(M3, 0, NM * 4, stream);
  hipMemsetAsync(rsy10, 0, (size_t)NBN0 * 4, stream);
  epass1<<<kB * E1 / 256, 256, 0, stream>>>(psm2, b1c, kN1, ptm2, b1r, kN0, rsy10, b1r, kN0, ebuf, E1);
  epass2<<<kB * E1 / 8, 256, 0, stream>>>(ebuf, rsy10, b1r, kN0, BIG, b1c, kN1, M3, E1);

  // z = relu(x0_out @ dec1_w + dec1_b), x0_out = relu(accy00)+relu(accy10)
  gemm_comp_br<<<NBN0 / 128, 256, 0, stream>>>(M2, M3, FULL, dec1w, dec1b, nullptr, 0, M1);

  // preds -> d_out (B, PRED, N0)
  decoder2<<<NBN0 / 256, 256, 0, stream>>>(M1, dec2w, dec2b, out);
}